// SparseSelfAttention_12919261626596
// MI455X (gfx1250) — compile-verified
//
#include <hip/hip_runtime.h>
#include <hip/hip_bf16.h>

// ---------------- problem constants ----------------
#define Bb 4
#define Ss 2048
#define Dd 1024
#define Ee 16
#define DH 64
#define TOPK 8
#define MAXLEN 1228            // int(S*0.6)
#define LPAD 1232              // 77*16, MAXLEN padded to multiple of 16
#define NTM 77                 // LPAD/16
#define CAPF 4.0f
#define EPSF 1e-6f

typedef __attribute__((ext_vector_type(2))) float v2f;
typedef __attribute__((ext_vector_type(8))) float v8f;

static __device__ __forceinline__ v8f vzero8() {
  v8f z;
  #pragma unroll
  for (int i = 0; i < 8; ++i) z[i] = 0.0f;
  return z;
}

// fp32 WMMA: D(16x16) = A(16x4) * B(4x16) + C.  Full fp32 precision.
static __device__ __forceinline__ v8f wmma_f32(v2f a, v2f b, v8f c) {
  return __builtin_amdgcn_wmma_f32_16x16x4_f32(false, a, false, b, (short)0, c,
                                               false, false);
}

// ---------------- workspace layout (floats) ----------------
// masked  : B*S*E          = 131072
// route   : E*B*S          = 131072
// seq_ids : E*B*MAXLEN int = 78592  (allotted 81920)
// Qw,Kw,Vw,ctx : 64*LPAD*64 = 5046272 each
#define OFF_MASKED 0
#define OFF_ROUTE  131072
#define OFF_SEQ    262144
#define OFF_Q      344064
#define PAIR_TOT   5046272   // 64*1232*64
// total = 344064 + 4*5046272 floats = ~78.3 MiB of d_ws

// ================= 1. gating: logits -> softmax -> top-8 mask =================
__global__ __launch_bounds__(256) void gate_kernel(const float* __restrict__ X,
                                                   const float* __restrict__ wg,
                                                   const float* __restrict__ bg,
                                                   float* __restrict__ masked) {
  const int token = blockIdx.x;          // b*S + s
  const int t = threadIdx.x;
  __shared__ float part[256][Ee];        // 16 KB
  float acc[Ee];
  #pragma unroll
  for (int e = 0; e < Ee; ++e) acc[e] = 0.0f;
  const float* xrow = X + (size_t)token * Dd;
  #pragma unroll
  for (int c = 0; c < 4; ++c) {
    const int d = t + 256 * c;
    const float xv = xrow[d];
    const float* wr = wg + (size_t)d * Ee;
    #pragma unroll
    for (int e = 0; e < Ee; ++e) acc[e] += xv * wr[e];
  }
  #pragma unroll
  for (int e = 0; e < Ee; ++e) part[t][e] = acc[e];
  __syncthreads();
  for (int s = 128; s > 0; s >>= 1) {
    if (t < s) {
      #pragma unroll
      for (int e = 0; e < Ee; ++e) part[t][e] += part[t + s][e];
    }
    __syncthreads();
  }
  if (t == 0) {
    float sc[Ee];
    float mx = -1e30f;
    #pragma unroll
    for (int e = 0; e < Ee; ++e) { sc[e] = part[0][e] + bg[e]; mx = fmaxf(mx, sc[e]); }
    float sum = 0.0f;
    #pragma unroll
    for (int e = 0; e < Ee; ++e) { sc[e] = expf(sc[e] - mx); sum += sc[e]; }
    const float inv = 1.0f / sum;
    #pragma unroll
    for (int e = 0; e < Ee; ++e) sc[e] *= inv;
    float* mo = masked + (size_t)token * Ee;
    #pragma unroll
    for (int e = 0; e < Ee; ++e) {
      int rank = 0;
      #pragma unroll
      for (int j = 0; j < Ee; ++j)
        rank += (sc[j] > sc[e]) || (sc[j] == sc[e] && j < e);
      mo[e] = (rank < TOPK) ? sc[e] : 0.0f;
    }
  }
}

// ================= 2. routing: denom over batch, capacity scale =================
__global__ __launch_bounds__(256) void route_kernel(const float* __restrict__ masked,
                                                    float* __restrict__ route) {
  const int i = blockIdx.x * 256 + threadIdx.x;   // < S*E
  if (i >= Ss * Ee) return;
  const int s = i / Ee, e = i % Ee;
  float m[Bb];
  float den = EPSF;
  #pragma unroll
  for (int b = 0; b < Bb; ++b) {
    m[b] = masked[((size_t)b * Ss + s) * Ee + e];
    den += m[b];
  }
  const float sc = CAPF / den;
  #pragma unroll
  for (int b = 0; b < Bb; ++b)
    route[((size_t)e * Bb + b) * Ss + s] = m[b] * sc;
}

// ===== 3. per (e,b): exact top-MAXLEN selection (lax.top_k tie semantics), sorted =====
__global__ __launch_bounds__(256) void select_kernel(const float* __restrict__ route,
                                                     int* __restrict__ seq) {
  const int pair = blockIdx.x;                    // e*B + b
  const float* r = route + (size_t)pair * Ss;
  __shared__ float vals[Ss];                      // 8 KB
  __shared__ unsigned char flag[Ss];
  __shared__ int csum[256];
  const int t = threadIdx.x;
  #pragma unroll
  for (int c = 0; c < 8; ++c) { const int i = t + 256 * c; vals[i] = r[i]; }
  __syncthreads();
  #pragma unroll
  for (int c = 0; c < 8; ++c) {
    const int i = t + 256 * c;
    const float vi = vals[i];
    int cnt = 0;
    for (int j = 0; j < Ss; ++j) {
      const float vj = vals[j];
      cnt += (vj > vi) || (vj == vi && j < i);
    }
    flag[i] = (cnt < MAXLEN) ? 1 : 0;
  }
  __syncthreads();
  int local = 0;
  #pragma unroll
  for (int c = 0; c < 8; ++c) local += flag[t * 8 + c];
  csum[t] = local;
  __syncthreads();
  if (t == 0) {
    int run = 0;
    for (int k = 0; k < 256; ++k) { const int v = csum[k]; csum[k] = run; run += v; }
  }
  __syncthreads();
  int base = csum[t];
  int* sq = seq + (size_t)pair * MAXLEN;
  #pragma unroll
  for (int c = 0; c < 8; ++c) {
    const int i = t * 8 + c;
    if (flag[i]) sq[base++] = i;
  }
}

// ===== 4. gathered QKV projection: (LPADx1024)@(1024x192) via fp32 WMMA =====
__global__ __launch_bounds__(128) void qkv_kernel(const float* __restrict__ X,
                                                  const float* __restrict__ Wq,
                                                  const int* __restrict__ seq,
                                                  float* __restrict__ Qw,
                                                  float* __restrict__ Kw,
                                                  float* __restrict__ Vw) {
  const int mt = blockIdx.x, pair = blockIdx.y;
  const int e = pair >> 2, b = pair & 3;
  __shared__ float As[16][33];                    // padded: conflict-free column reads
  __shared__ float Bs[32][200];                   // 192 padded to 200
  const int t = threadIdx.x;
  const int lane = t & 31, w = t >> 5;
  const int m = lane & 15, h = lane >> 4, n = m;
  const int m0 = mt * 16;
  const int* sq = seq + (size_t)pair * MAXLEN;
  const float* Wqe = Wq + (size_t)e * Dd * 192;
  // hoist gather rows for this thread's A-load slots
  int arow[4]; int acol[4]; const float* asrc[4];
  #pragma unroll
  for (int c = 0; c < 4; ++c) {
    const int i = t + 128 * c;
    arow[c] = i >> 5; acol[c] = i & 31;
    const int mg = m0 + arow[c];
    asrc[c] = (mg < MAXLEN) ? (X + ((size_t)b * Ss + sq[mg]) * Dd) : nullptr;
  }
  v8f acc0 = vzero8(), acc1 = vzero8(), acc2 = vzero8();
  for (int k0 = 0; k0 < Dd; k0 += 32) {
    #pragma unroll
    for (int c = 0; c < 4; ++c)
      As[arow[c]][acol[c]] = asrc[c] ? asrc[c][k0 + acol[c]] : 0.0f;
    #pragma unroll
    for (int c = 0; c < 48; ++c) {
      const int i = t + 128 * c;
      const int r = i / 192, col = i % 192;
      Bs[r][col] = Wqe[(size_t)(k0 + r) * 192 + col];
    }
    __syncthreads();
    const int nb = w * 48 + n;
    #pragma unroll
    for (int ks = 0; ks < 8; ++ks) {
      const int k = ks * 4;
      v2f a; a.x = As[m][k + 2 * h]; a.y = As[m][k + 2 * h + 1];
      v2f b0, b1, b2;
      b0.x = Bs[k + 2 * h][nb];      b0.y = Bs[k + 2 * h + 1][nb];
      b1.x = Bs[k + 2 * h][nb + 16]; b1.y = Bs[k + 2 * h + 1][nb + 16];
      b2.x = Bs[k + 2 * h][nb + 32]; b2.y = Bs[k + 2 * h + 1][nb + 32];
      acc0 = wmma_f32(a, b0, acc0);
      acc1 = wmma_f32(a, b1, acc1);
      acc2 = wmma_f32(a, b2, acc2);
    }
    __syncthreads();
  }
  const size_t rowBase = (size_t)pair * LPAD + m0;
  #pragma unroll
  for (int tt = 0; tt < 3; ++tt) {
    const v8f acc = (tt == 0) ? acc0 : (tt == 1) ? acc1 : acc2;
    const int f0 = w * 48 + tt * 16;              // tile never straddles a 64-boundary
    float* dst = (f0 < 64) ? Qw : (f0 < 128) ? Kw : Vw;
    const int dcol = (f0 & 63) + n;
    #pragma unroll
    for (int r = 0; r < 8; ++r)
      dst[(rowBase + r + 8 * h) * DH + dcol] = acc[r];
  }
}

// ===== 5. causal flash attention per (pair, 16-row tile); one wave32 =====
__global__ __launch_bounds__(32) void attn_kernel(const float* __restrict__ Qw,
                                                  const float* __restrict__ Kw,
                                                  const float* __restrict__ Vw,
                                                  float* __restrict__ ctx) {
  const int mt = blockIdx.x, pair = blockIdx.y;
  __shared__ float Qs[16][65];
  __shared__ float Ks[16][65];
  __shared__ float Vs[16][72];
  __shared__ float Ps[16][17];
  __shared__ float mrow[16], lrow[16];
  const int t = threadIdx.x;
  const int m = t & 15, h = t >> 4, n = m;
  const size_t pb = (size_t)pair * LPAD;
  const int m0 = mt * 16;
  #pragma unroll
  for (int c = 0; c < 32; ++c) {
    const int i = t + 32 * c;
    const int r = i >> 6, col = i & 63;
    Qs[r][col] = Qw[(pb + m0 + r) * DH + col];
  }
  if (t < 16) { mrow[t] = -1e30f; lrow[t] = 0.0f; }
  v8f o[4]; o[0] = vzero8(); o[1] = vzero8(); o[2] = vzero8(); o[3] = vzero8();
  __syncthreads();
  for (int jt = 0; jt <= mt; ++jt) {
    #pragma unroll
    for (int c = 0; c < 32; ++c) {
      const int i = t + 32 * c;
      const int r = i >> 6, col = i & 63;
      Ks[r][col] = Kw[(pb + jt * 16 + r) * DH + col];
      Vs[r][col] = Vw[(pb + jt * 16 + r) * DH + col];
    }
    __syncthreads();
    v8f sacc = vzero8();
    #pragma unroll
    for (int ks = 0; ks < 16; ++ks) {
      const int k = 4 * ks;
      v2f a;  a.x = Qs[m][k + 2 * h];  a.y = Qs[m][k + 2 * h + 1];
      v2f bv; bv.x = Ks[n][k + 2 * h]; bv.y = Ks[n][k + 2 * h + 1];
      sacc = wmma_f32(a, bv, sacc);
    }
    // scale 1/sqrt(64), causal mask (col>row) and col>=MAXLEN mask
    const int gj = jt * 16 + n;
    float sv[8], pv[8], mnew[8], fac[8], lsum[8], lold[8];
    #pragma unroll
    for (int r = 0; r < 8; ++r) {
      const int gi = m0 + r + 8 * h;
      float v = sacc[r] * 0.125f;
      if (gj > gi || gj >= MAXLEN) v = -1e30f;
      sv[r] = v;
    }
    #pragma unroll
    for (int r = 0; r < 8; ++r) {
      float red = sv[r];
      red = fmaxf(red, __shfl_xor(red, 1, 16));
      red = fmaxf(red, __shfl_xor(red, 2, 16));
      red = fmaxf(red, __shfl_xor(red, 4, 16));
      red = fmaxf(red, __shfl_xor(red, 8, 16));
      const float old = mrow[r + 8 * h];
      const float nm = fmaxf(old, red);
      mnew[r] = nm;
      fac[r] = expf(old - nm);
      pv[r] = expf(sv[r] - nm);
    }
    #pragma unroll
    for (int r = 0; r < 8; ++r) {
      float red = pv[r];
      red += __shfl_xor(red, 1, 16);
      red += __shfl_xor(red, 2, 16);
      red += __shfl_xor(red, 4, 16);
      red += __shfl_xor(red, 8, 16);
      lsum[r] = red;
      lold[r] = lrow[r + 8 * h];
    }
    __syncthreads();
    if (n == 0) {
      #pragma unroll
      for (int r = 0; r < 8; ++r) {
        mrow[r + 8 * h] = mnew[r];
        lrow[r + 8 * h] = lold[r] * fac[r] + lsum[r];
      }
    }
    #pragma unroll
    for (int r = 0; r < 8; ++r) Ps[r + 8 * h][n] = pv[r];
    #pragma unroll
    for (int nt = 0; nt < 4; ++nt)
      #pragma unroll
      for (int r = 0; r < 8; ++r) o[nt][r] *= fac[r];
    __syncthreads();
    #pragma unroll
    for (int nt = 0; nt < 4; ++nt) {
      #pragma unroll
      for (int ks = 0; ks < 4; ++ks) {
        const int k = 4 * ks;
        v2f a;  a.x = Ps[m][k + 2 * h];           a.y = Ps[m][k + 2 * h + 1];
        v2f bv; bv.x = Vs[k + 2 * h][nt * 16 + n]; bv.y = Vs[k + 2 * h + 1][nt * 16 + n];
        o[nt] = wmma_f32(a, bv, o[nt]);
      }
    }
    __syncthreads();
  }
  float linv[8];
  #pragma unroll
  for (int r = 0; r < 8; ++r) linv[r] = 1.0f / lrow[r + 8 * h];
  #pragma unroll
  for (int nt = 0; nt < 4; ++nt)
    #pragma unroll
    for (int r = 0; r < 8; ++r)
      ctx[(pb + m0 + r + 8 * h) * DH + nt * 16 + n] = o[nt][r] * linv[r];
}

// ===== 6. FF projection + bias, scatter-add into residual via f32 atomics =====
__global__ __launch_bounds__(128) void ff_kernel(const float* __restrict__ ctx,
                                                 const float* __restrict__ Wf,
                                                 const float* __restrict__ bff,
                                                 const int* __restrict__ seq,
                                                 float* __restrict__ y) {
  const int mt = blockIdx.x, pair = blockIdx.y;
  const int e = pair >> 2, b = pair & 3;
  __shared__ float As[16][65];
  const int t = threadIdx.x;
  const int lane = t & 31, w = t >> 5;
  const int m = lane & 15, h = lane >> 4, n = m;
  const int m0 = mt * 16;
  const size_t rowBase = (size_t)pair * LPAD + m0;
  #pragma unroll
  for (int c = 0; c < 8; ++c) {
    const int i = t + 128 * c;
    const int r = i >> 6, col = i & 63;
    As[r][col] = ctx[(rowBase + r) * DH + col];
  }
  __syncthreads();
  const float* We = Wf + (size_t)e * DH * Dd;
  const int* sq = seq + (size_t)pair * MAXLEN;
  int drow[8];
  #pragma unroll
  for (int r = 0; r < 8; ++r) {
    const int mg = m0 + r + 8 * h;
    drow[r] = (mg < MAXLEN) ? sq[mg] : -1;
  }
  for (int nt = w; nt < 64; nt += 4) {
    const int n0 = nt * 16 + n;
    v8f acc = vzero8();
    #pragma unroll
    for (int ks = 0; ks < 16; ++ks) {
      const int k = 4 * ks;
      v2f a;  a.x = As[m][k + 2 * h]; a.y = As[m][k + 2 * h + 1];
      v2f bv;
      const float* wp = We + (size_t)(k + 2 * h) * Dd + n0;
      __builtin_prefetch(wp + 64, 0, 1);          // global_prefetch for next n-tile
      bv.x = wp[0]; bv.y = wp[Dd];
      acc = wmma_f32(a, bv, acc);
    }
    const float bias = bff[(size_t)e * Dd + n0];
    #pragma unroll
    for (int r = 0; r < 8; ++r)
      if (drow[r] >= 0)
        atomicAdd(&y[((size_t)b * Ss + drow[r]) * Dd + n0], acc[r] + bias);
  }
}

// ================= 7. LayerNorm (in place on d_out) =================
__global__ __launch_bounds__(256) void ln_kernel(float* __restrict__ y,
                                                 const float* __restrict__ gamma,
                                                 const float* __restrict__ beta) {
  const int token = blockIdx.x;
  const int t = threadIdx.x;
  __shared__ float red[256], red2[256];
  float v[4]; float s = 0.0f, s2 = 0.0f;
  float* row = y + (size_t)token * Dd;
  #pragma unroll
  for (int c = 0; c < 4; ++c) {
    v[c] = row[t + 256 * c];
    s += v[c]; s2 += v[c] * v[c];
  }
  red[t] = s; red2[t] = s2;
  __syncthreads();
  for (int st = 128; st > 0; st >>= 1) {
    if (t < st) { red[t] += red[t + st]; red2[t] += red2[t + st]; }
    __syncthreads();
  }
  const float mu = red[0] * (1.0f / Dd);
  const float var = red2[0] * (1.0f / Dd) - mu * mu;
  const float inv = rsqrtf(var + 1e-5f);
  #pragma unroll
  for (int c = 0; c < 4; ++c) {
    const int d = t + 256 * c;
    row[d] = (v[c] - mu) * inv * gamma[d] + beta[d];
  }
}

// ================= host launch =================
extern "C" void kernel_launch(void* const* d_in, const int* in_sizes, int n_in,
                              void* d_out, int out_size, void* d_ws, size_t ws_size,
                              hipStream_t stream) {
  const float* X  = (const float*)d_in[0];
  // d_in[1] = attn_mask (bool, recomputed analytically on device)
  const float* wg = (const float*)d_in[2];
  const float* bg = (const float*)d_in[3];
  const float* Wq = (const float*)d_in[4];
  const float* Wf = (const float*)d_in[5];
  const float* bf = (const float*)d_in[6];
  const float* ga = (const float*)d_in[7];
  const float* be = (const float*)d_in[8];
  float* y = (float*)d_out;

  float* w = (float*)d_ws;
  float* masked = w + OFF_MASKED;
  float* route  = w + OFF_ROUTE;
  int*   seq    = (int*)(w + OFF_SEQ);
  float* Qw     = w + OFF_Q;
  float* Kw     = Qw + PAIR_TOT;
  float* Vw     = Kw + PAIR_TOT;
  float* ctx    = Vw + PAIR_TOT;

  // residual seed: y = X
  hipMemcpyAsync(y, X, (size_t)Bb * Ss * Dd * sizeof(float),
                 hipMemcpyDeviceToDevice, stream);

  gate_kernel<<<Bb * Ss, 256, 0, stream>>>(X, wg, bg, masked);
  route_kernel<<<(Ss * Ee + 255) / 256, 256, 0, stream>>>(masked, route);
  select_kernel<<<Ee * Bb, 256, 0, stream>>>(route, seq);
  qkv_kernel<<<dim3(NTM, Ee * Bb), 128, 0, stream>>>(X, Wq, seq, Qw, Kw, Vw);
  attn_kernel<<<dim3(NTM, Ee * Bb), 32, 0, stream>>>(Qw, Kw, Vw, ctx);
  ff_kernel<<<dim3(NTM, Ee * Bb), 128, 0, stream>>>(ctx, Wf, bf, seq, y);
  ln_kernel<<<Bb * Ss, 256, 0, stream>>>(y, ga, be);
}